// knn_70824010711496
// MI455X (gfx1250) — compile-verified
//
#include <hip/hip_runtime.h>
#include <hip/hip_bf16.h>

// Gather: out[b,n,k,:] = features[b, topk[b,n,k], :]
// B=32, N=4096, K=20, F=64 (fp32).
//
// Memory-bound: 671 MB output stream, 33.5 MB feature working set (L2-resident
// on MI455X's 192 MB L2). Strategy:
//   - one thread per float4 chunk (16 lanes per 256B row) -> global_load_b128 /
//     global_store_b128, fully coalesced under wave32
//   - NT store hint on the output stream so it doesn't evict features from L2
//   - NT load hint on indices (single use); RT loads on features (keep in L2)

#define KNN_B 32
#define KNN_N 4096
#define KNN_K 20
#define KNN_F 64

typedef float f4 __attribute__((ext_vector_type(4)));

__global__ __launch_bounds__(256) void knn_gather_kernel(
    const int* __restrict__ topk,      // [B, N, K]
    const float* __restrict__ feat,    // [B, N, F]
    float* __restrict__ out)           // [B, N, K, F]
{
    // 16 float4 chunks per row; rows = B*N*K = 2,621,440; threads = rows*16.
    const int gid = blockIdx.x * 256 + threadIdx.x;
    const int total = KNN_B * KNN_N * KNN_K * (KNN_F / 4);
    if (gid >= total) return;

    const int row = gid >> 4;          // flat (b, n, k)
    const int c   = gid & 15;          // which float4 of the 64-float row
    const int b   = row / (KNN_N * KNN_K);

    // Index is consumed exactly once per half-wave group -> non-temporal load.
    const int idx = __builtin_nontemporal_load(topk + row);

    // Feature row: keep temporal (RT) so the 33.5 MB array stays L2-resident.
    const f4* __restrict__ src =
        (const f4*)(feat + (size_t)(b * KNN_N + idx) * KNN_F) + c;
    f4 v = *src;

    // Output is a 671 MB pure stream: NT store so it rinses through L2
    // without evicting the feature working set.
    f4* __restrict__ dst = (f4*)(out + (size_t)row * KNN_F) + c;
    __builtin_nontemporal_store(v, dst);
}

extern "C" void kernel_launch(void* const* d_in, const int* in_sizes, int n_in,
                              void* d_out, int out_size, void* d_ws, size_t ws_size,
                              hipStream_t stream) {
    const int*   topk = (const int*)d_in[0];    // [32, 4096, 20] int32
    const float* feat = (const float*)d_in[1];  // [32, 4096, 64] float32
    float*       out  = (float*)d_out;          // [32, 4096, 20, 64] float32

    const int total  = KNN_B * KNN_N * KNN_K * (KNN_F / 4);  // 41,943,040 threads
    const int blocks = (total + 255) / 256;                  // 163,840 blocks

    knn_gather_kernel<<<blocks, 256, 0, stream>>>(topk, feat, out);
}